// QLoRa_83623013253769
// MI455X (gfx1250) — compile-verified
//
#include <hip/hip_runtime.h>

typedef __attribute__((ext_vector_type(2))) float v2f;
typedef __attribute__((ext_vector_type(8))) float v8f;

#define D        4096
#define NTOK     16384            // BATCH * SEQ
#define KCH      64               // K-chunk per buffer
#define XPITCH   68               // 64 + 4 pad: conflict-free strided reads
#define WAVES    2
#define TOKW     16               // tokens per wave (WMMA M)
#define TOKB     (WAVES * TOKW)

// A' LDS layout: row0 @ 0, row1 @ D+4, zero row @ 2D+8 (bank-staggered)
#define AP_R1    (D + 4)
#define AP_Z     (2 * D + 8)
#define AP_SZ    (2 * D + 8 + 64)

__device__ __forceinline__ float s8f(unsigned v) {
    return (float)(int)(signed char)(v & 0xFFu);
}

__global__ __launch_bounds__(64) void qlora_fused(
    const float* __restrict__ x,
    const signed char* __restrict__ aq,      // [2][4096] int8
    const float* __restrict__ ascale,        // [2]
    const signed char* __restrict__ bq,      // [4096][2] int8
    const float* __restrict__ bscale,        // [4096]
    float* __restrict__ out)
{
    __shared__ float s_ap[AP_SZ];                      // dequant A' + zero row
    __shared__ float s_x[WAVES][2][TOKW * XPITCH];     // double-buffered X tiles
    __shared__ float s_h[WAVES][TOKW][2];              // h, pre-scaled by 0.5

    const int tid  = threadIdx.x;
    const int wave = tid >> 5;
    const int lane = tid & 31;

    // ---- one-time: dequantize A' into LDS (row1 shifted +4, zero row) ----
    {
        const int* aq4 = (const int*)aq;               // 2048 dwords
        const float sc0 = ascale[0], sc1 = ascale[1];
#pragma unroll
        for (int i = 0; i < 32; ++i) {
            int j   = i * 64 + tid;
            int v   = aq4[j];
            int idx = j * 4;
            int dst = idx + ((idx < D) ? 0 : 4);       // row1 staggered by +4
            float s = (idx < D) ? sc0 : sc1;
            s_ap[dst + 0] = s8f((unsigned)v)       * s;
            s_ap[dst + 1] = s8f((unsigned)v >> 8)  * s;
            s_ap[dst + 2] = s8f((unsigned)v >> 16) * s;
            s_ap[dst + 3] = s8f((unsigned)v >> 24) * s;
        }
        s_ap[AP_Z + tid] = 0.0f;                       // 64-float zero row
    }
    __syncthreads();

    const int tok0 = blockIdx.x * TOKB + wave * TOKW;
    const float* xb = x + (size_t)tok0 * D;

    const int mrow = lane & 15;      // A-frag token row M ; B-frag rank column N
    const int half = lane >> 4;
    const int kk   = 2 * half;       // lane holds K = {kk, kk+1} of each x4 step
    const int col  = mrow * 4;       // staging column (float4 granules)

    // LDS byte address base of this wave's X buffers (generic ptr low 32 bits)
    const unsigned xlds0 = (unsigned)(uintptr_t)&s_x[wave][0][0];
    const unsigned xldsB = (unsigned)(TOKW * XPITCH * 4);   // bytes per buffer
    const unsigned stg   = (unsigned)((half * XPITCH + col) * 4);

    // issue 8 async global->LDS b128 loads for one 16x64 chunk (ASYNCcnt += 8)
    auto issue_chunk = [&](int kc, int b) {
        const float* gp = xb + (size_t)half * D + kc + col;
        unsigned     lp = xlds0 + (unsigned)b * xldsB + stg;
#pragma unroll
        for (int i = 0; i < 8; ++i) {   // 2 rows per issue: r = 2*i + half
            asm volatile("global_load_async_to_lds_b128 %0, %1, off"
                         :: "v"(lp), "v"(gp) : "memory");
            gp += (size_t)2 * D;
            lp += (unsigned)(2 * XPITCH * 4);
        }
    };

    // 16 WMMA accumulation steps over one resident 16x64 buffer
    v8f c = {};
    auto wmma_chunk = [&](int kc, int b) {
        // branchless per-lane B base: row0 / row1(+4) / zero row (no EXEC ops)
        int boff = (mrow == 0) ? (kc + kk)
                 : (mrow == 1) ? (AP_R1 + kc + kk)
                                : (AP_Z + kk);
        const float* bp = s_ap + boff;
        const float* ap = &s_x[wave][b][mrow * XPITCH + kk];
#pragma unroll
        for (int k = 0; k < KCH; k += 4) {
            v2f a = *(const v2f*)(ap + k);             // X[mrow][k+kk], +1
            v2f bf = *(const v2f*)(bp + k);            // A'[N][k+kk] or zeros
            c = __builtin_amdgcn_wmma_f32_16x16x4_f32(
                    false, a, false, bf, (short)0, c, false, false);
        }
    };

    // ---- phase 1: H[16x2] = X[16x4096] @ A'^T, software-pipelined ----
    issue_chunk(0, 0);
    int buf = 0;
#pragma unroll 1
    for (int kc = 0; kc < D - KCH; kc += KCH) {        // steady state: no branches
        issue_chunk(kc + KCH, buf ^ 1);                // stream next buffer
        asm volatile("s_wait_asynccnt 0x8" ::: "memory");   // cur buffer resident
        wmma_chunk(kc, buf);
        buf ^= 1;
    }
    asm volatile("s_wait_asynccnt 0x0" ::: "memory");  // peeled last chunk
    wmma_chunk(D - KCH, buf);

    // ---- hand off: C[m][n] cols 0,1 -> LDS, fold alpha/rank = 0.5 ----
    // C layout: lane N = lane&15 ; VGPR i holds M = i + 8*(lane>>4)
    if (mrow < 2) {
#pragma unroll
        for (int i = 0; i < 8; ++i)
            s_h[wave][i + 8 * half][mrow] = c[i] * 0.5f;
    }

    float h0[TOKW], h1[TOKW];
#pragma unroll
    for (int t = 0; t < TOKW; ++t) {
        v2f hv = *(const v2f*)&s_h[wave][t][0];
        h0[t] = hv.x;
        h1[t] = hv.y;
    }

    // ---- phase 2: out[16 x 4096] = h @ B'^T (rank-2 expansion, store-bound) ----
#pragma unroll 1
    for (int dc = 0; dc < D; dc += 128) {
        int d0 = dc + lane * 4;
        unsigned long long raw = *(const unsigned long long*)(bq + (size_t)2 * d0);
        float4 sc = *(const float4*)(bscale + d0);
        float B0x = s8f((unsigned)(raw      )) * sc.x;
        float B1x = s8f((unsigned)(raw >>  8)) * sc.x;
        float B0y = s8f((unsigned)(raw >> 16)) * sc.y;
        float B1y = s8f((unsigned)(raw >> 24)) * sc.y;
        float B0z = s8f((unsigned)(raw >> 32)) * sc.z;
        float B1z = s8f((unsigned)(raw >> 40)) * sc.z;
        float B0w = s8f((unsigned)(raw >> 48)) * sc.w;
        float B1w = s8f((unsigned)(raw >> 56)) * sc.w;

        float* op = out + (size_t)tok0 * D + d0;
#pragma unroll
        for (int t = 0; t < TOKW; ++t) {
            float4 o;
            o.x = h0[t] * B0x + h1[t] * B1x;
            o.y = h0[t] * B0y + h1[t] * B1y;
            o.z = h0[t] * B0z + h1[t] * B1z;
            o.w = h0[t] * B0w + h1[t] * B1w;
            *(float4*)(op + (size_t)t * D) = o;
        }
    }
}

extern "C" void kernel_launch(void* const* d_in, const int* in_sizes, int n_in,
                              void* d_out, int out_size, void* d_ws, size_t ws_size,
                              hipStream_t stream) {
    const float*       x      = (const float*)d_in[0];
    const signed char* aq     = (const signed char*)d_in[1];
    const float*       ascale = (const float*)d_in[2];
    const signed char* bq     = (const signed char*)d_in[3];
    const float*       bscale = (const float*)d_in[4];
    float*             outp   = (float*)d_out;

    dim3 grid(NTOK / TOKB);   // 512 blocks
    dim3 block(64);           // 2 wave32s
    qlora_fused<<<grid, block, 0, stream>>>(x, aq, ascale, bq, bscale, outp);
}